// VFE_Block_54889682043257
// MI455X (gfx1250) — compile-verified
//
#include <hip/hip_runtime.h>
#include <hip/hip_bf16.h>

typedef float v2f __attribute__((ext_vector_type(2)));
typedef float v8f __attribute__((ext_vector_type(8)));

#define BN_EPS 1e-3f

// ---------------------------------------------------------------------------
// Kernel 0: zero the dense output grid (1.44 GB) with vectorized stores.
// ---------------------------------------------------------------------------
__global__ void zero_f4(float* __restrict__ out, long long n) {
    long long i = (long long)blockIdx.x * blockDim.x + threadIdx.x;
    long long stride = (long long)gridDim.x * blockDim.x;
    long long n4 = n >> 2;
    float4 z = make_float4(0.f, 0.f, 0.f, 0.f);
    for (long long j = i; j < n4; j += stride)
        reinterpret_cast<float4*>(out)[j] = z;
    long long tail = n & 3;
    if (i < tail) out[n - 1 - i] = 0.f;
}

// ---------------------------------------------------------------------------
// Kernel 1: fully fused VFE block, one 128-thread block (4 wave32) per voxel.
// ---------------------------------------------------------------------------
__global__ __launch_bounds__(128)
void vfe_kernel(const float* __restrict__ x,        // [B,K,35,7]
                const int*   __restrict__ coords,   // [B,K,4]
                const long long* __restrict__ shape,// [B,D,H,W,C] int64
                const float* __restrict__ w1, const float* __restrict__ b1,
                const float* __restrict__ g1, const float* __restrict__ beta1,
                const float* __restrict__ mu1, const float* __restrict__ var1,
                const float* __restrict__ w2, const float* __restrict__ b2,
                const float* __restrict__ g2, const float* __restrict__ beta2,
                const float* __restrict__ mu2, const float* __restrict__ var2,
                const float* __restrict__ wf, const float* __restrict__ bfv,
                float* __restrict__ out)            // [B,C,D,H,W]
{
    constexpr int T = 35;    // points per voxel
    constexpr int TP = 48;   // padded to 3 WMMA row-tiles
    constexpr int F = 7;

    __shared__ float h1[TP][33];    // layer-1 output (32 cols, +1 pad)
    __shared__ float h2[TP][129];   // layer-2 output (128 cols, +1 pad)
    __shared__ float mp[64];        // column maxes (16 used for L1, 64 for L2)
    __shared__ float maskS[TP];

    const int tid  = threadIdx.x;
    const int lane = tid & 31;
    const int wave = tid >> 5;
    const int bk   = blockIdx.x;

    // ----------------- Stage 1: dense 7->16 + BN (scalar, tiny) -----------
    if (tid < TP) {
        const int t = tid;
        if (t < T) {
            const float* xp = x + ((size_t)bk * T + t) * F;
            float xr[F];
            float mx = -3.4e38f;
            #pragma unroll
            for (int f = 0; f < F; ++f) { xr[f] = xp[f]; mx = fmaxf(mx, xr[f]); }
            maskS[t] = (mx != 0.0f) ? 1.0f : 0.0f;
            #pragma unroll
            for (int u = 0; u < 16; ++u) {
                float s = b1[u];
                #pragma unroll
                for (int f = 0; f < F; ++f) s = fmaf(xr[f], w1[f * 16 + u], s);
                s = fmaxf(s, 0.0f);
                float sc = g1[u] * rsqrtf(var1[u] + BN_EPS);
                h1[t][u] = (s - mu1[u]) * sc + beta1[u];   // pre-mask
            }
        } else {
            maskS[t] = 0.0f;
        }
    }
    __syncthreads();

    if (tid < 16) {                       // column max over valid T rows
        float m = -3.4e38f;
        for (int t = 0; t < T; ++t) m = fmaxf(m, h1[t][tid]);
        mp[tid] = m;
    }
    __syncthreads();

    if (tid < TP) {                       // masked concat -> h1[48][32]
        const float msk = maskS[tid];
        const bool valid = (tid < T);
        #pragma unroll
        for (int u = 0; u < 16; ++u) {
            h1[tid][u]      = valid ? h1[tid][u] * msk : 0.0f;
            h1[tid][16 + u] = valid ? mp[u] * msk      : 0.0f;
        }
    }
    __syncthreads();

    // ----------------- Stage 2: WMMA h1(48x32) x w2(32x64) ----------------
    {
        const int nt   = wave;                     // N-tile: cols nt*16..+15
        const int mcol = lane & 15;
        const int khalf = (lane >> 4) * 2;         // A/B K split across lane halves
        const int bcol = nt * 16 + mcol;

        for (int mtile = 0; mtile < 3; ++mtile) {
            v8f acc = {0.f, 0.f, 0.f, 0.f, 0.f, 0.f, 0.f, 0.f};
            const int arow = mtile * 16 + mcol;
            #pragma unroll
            for (int kk = 0; kk < 8; ++kk) {       // K = 32, 4 per WMMA
                v2f a, bb;
                a[0]  = h1[arow][kk * 4 + khalf + 0];
                a[1]  = h1[arow][kk * 4 + khalf + 1];
                bb[0] = w2[(kk * 4 + khalf + 0) * 64 + bcol];
                bb[1] = w2[(kk * 4 + khalf + 1) * 64 + bcol];
                acc = __builtin_amdgcn_wmma_f32_16x16x4_f32(
                        false, a, false, bb, (short)0, acc, false, false);
            }
            // epilogue: bias + relu + BN2, write to h2 cols [0,64)
            const float b2c  = b2[bcol];
            const float s2c  = g2[bcol] * rsqrtf(var2[bcol] + BN_EPS);
            const float mu2c = mu2[bcol];
            const float be2c = beta2[bcol];
            const int rbase = mtile * 16 + (lane >> 4) * 8;
            #pragma unroll
            for (int r = 0; r < 8; ++r) {
                float v = fmaxf(acc[r] + b2c, 0.0f);
                h2[rbase + r][bcol] = (v - mu2c) * s2c + be2c;
            }
        }
    }
    __syncthreads();

    if (tid < 64) {                       // column max over valid T rows
        float m = -3.4e38f;
        for (int t = 0; t < T; ++t) m = fmaxf(m, h2[t][tid]);
        mp[tid] = m;
    }
    __syncthreads();

    for (int idx = tid; idx < TP * 128; idx += 128) {  // masked concat rewrite
        const int t = idx >> 7;
        const int c = idx & 127;
        const float msk = maskS[t];
        const bool valid = (t < T);
        float v;
        if (c < 64) v = valid ? h2[t][c] * msk  : 0.0f;
        else        v = valid ? mp[c - 64] * msk : 0.0f;
        h2[t][c] = v;
    }
    __syncthreads();

    // ----------------- Stage 3: WMMA h2(48x128) x wf(128x128) -------------
    const long long Dd = shape[1], Hh = shape[2], Ww = shape[3], Cc = shape[4];
    const int cb = coords[bk * 4 + 0];
    const int cd = coords[bk * 4 + 1];
    const int ch = coords[bk * 4 + 2];
    const int cw = coords[bk * 4 + 3];
    const long long spatial = (long long)cd * (Hh * Ww) + (long long)ch * Ww + cw;
    const long long DHW = Dd * Hh * Ww;

    const int mcol  = lane & 15;
    const int khalf = (lane >> 4) * 2;

    for (int i = 0; i < 2; ++i) {
        const int nt   = wave * 2 + i;             // 8 N-tiles over 4 waves
        const int bcol = nt * 16 + mcol;
        const float bfc = bfv[bcol];
        float vmax = 0.0f;                         // relu output >= 0

        for (int mtile = 0; mtile < 3; ++mtile) {
            v8f acc = {0.f, 0.f, 0.f, 0.f, 0.f, 0.f, 0.f, 0.f};
            const int arow = mtile * 16 + mcol;
            #pragma unroll
            for (int kk = 0; kk < 32; ++kk) {      // K = 128, 4 per WMMA
                v2f a, bb;
                a[0]  = h2[arow][kk * 4 + khalf + 0];
                a[1]  = h2[arow][kk * 4 + khalf + 1];
                bb[0] = wf[(kk * 4 + khalf + 0) * 128 + bcol];
                bb[1] = wf[(kk * 4 + khalf + 1) * 128 + bcol];
                acc = __builtin_amdgcn_wmma_f32_16x16x4_f32(
                        false, a, false, bb, (short)0, acc, false, false);
            }
            #pragma unroll
            for (int r = 0; r < 8; ++r)
                vmax = fmaxf(vmax, fmaxf(acc[r] + bfc, 0.0f));
        }
        // fold the two row-halves (lane L and L+16 hold the same column)
        float other = __shfl_xor(vmax, 16, 32);
        vmax = fmaxf(vmax, other);

        if (lane < 16) {
            long long oidx = ((long long)cb * Cc + bcol) * DHW + spatial;
            atomicAdd(&out[oidx], vmax);           // scatter-add (dup coords sum)
        }
    }
}

// ---------------------------------------------------------------------------
extern "C" void kernel_launch(void* const* d_in, const int* in_sizes, int n_in,
                              void* d_out, int out_size, void* d_ws, size_t ws_size,
                              hipStream_t stream) {
    (void)n_in; (void)d_ws; (void)ws_size;

    const float*     x      = (const float*)d_in[0];
    const int*       coords = (const int*)d_in[1];
    const long long* shape  = (const long long*)d_in[2];
    const float* w1 = (const float*)d_in[3],  *b1 = (const float*)d_in[4];
    const float* g1 = (const float*)d_in[5],  *beta1 = (const float*)d_in[6];
    const float* mu1 = (const float*)d_in[7], *var1 = (const float*)d_in[8];
    const float* w2 = (const float*)d_in[9],  *b2 = (const float*)d_in[10];
    const float* g2 = (const float*)d_in[11], *beta2 = (const float*)d_in[12];
    const float* mu2 = (const float*)d_in[13],*var2 = (const float*)d_in[14];
    const float* wf = (const float*)d_in[15], *bf = (const float*)d_in[16];
    float* out = (float*)d_out;

    const long long n = (long long)out_size;           // 360,448,000 fp32
    const int BK = in_sizes[1] / 4;                    // B*K voxels (24000)

    // 1) zero the dense grid (bandwidth-bound: ~1.44 GB of stores)
    long long n4 = (n + 3) >> 2;
    unsigned zblocks = (unsigned)((n4 + 255) / 256);
    if (zblocks == 0) zblocks = 1;
    zero_f4<<<zblocks, 256, 0, stream>>>(out, n);

    // 2) fused VFE + scatter-add, one block per voxel
    vfe_kernel<<<BK, 128, 0, stream>>>(x, coords, shape,
                                       w1, b1, g1, beta1, mu1, var1,
                                       w2, b2, g2, beta2, mu2, var2,
                                       wf, bf, out);
}